// DerivativeNet_29592324670188
// MI455X (gfx1250) — compile-verified
//
#include <hip/hip_runtime.h>
#include <stdint.h>

// DerivativeNet: masked 1-D finite difference along last axis.
// u: (8,4,1024,1024) f32, nmask: (1,1,1024,1024) f32 broadcast over (b,c).
// Memory-bound (~272 MB @ 23.3 TB/s ~ 12us). Strategy: CDNA5 async
// global->LDS staging (ASYNCcnt path) of each row + mask row with zeroed
// halo, compute from LDS, non-temporal 128b stores for the write-once output.
// Cache policy: u is streamed once -> TH_LOAD_NT; mask is reused 32x and
// fits in L2 (4MB << 192MB) -> TH_LOAD_HT; output write-once -> NT store.

typedef float v4f __attribute__((ext_vector_type(4)));

#define ROW_W 1024

__global__ __launch_bounds__(256) void DerivativeNet_kernel(
    const float* __restrict__ u,
    const float* __restrict__ nmask,
    float* __restrict__ out)
{
    __shared__ float s_u[ROW_W + 2];
    __shared__ float s_m[ROW_W + 2];

    const int tid = threadIdx.x;
    const int row = blockIdx.x;             // flat (b,c,y) row index
    const int y   = row & (ROW_W - 1);      // mask row (broadcast over b,c)
    const int i   = tid * 4;                // first element this thread owns

    const float* u_row = u     + (size_t)row * ROW_W;
    const float* m_row = nmask + (size_t)y   * ROW_W;
    float*       o_row = out   + (size_t)row * ROW_W;

    // --- Stage row + mask into LDS via CDNA5 async global->LDS copies ---
    // 256 lanes x 16B = 4KB = one full row per instruction "sweep".
    {
        uint32_t lds_u = (uint32_t)(uintptr_t)&s_u[1 + i];
        uint32_t lds_m = (uint32_t)(uintptr_t)&s_m[1 + i];
        uint64_t ga_u  = (uint64_t)(uintptr_t)(u_row + i);
        uint64_t ga_m  = (uint64_t)(uintptr_t)(m_row + i);
        // u: streamed once, don't pollute L2
        asm volatile("global_load_async_to_lds_b128 %0, %1, off th:TH_LOAD_NT"
                     :: "v"(lds_u), "v"(ga_u) : "memory");
        // mask: reused by 32 (b,c) slices, keep hot in L2
        asm volatile("global_load_async_to_lds_b128 %0, %1, off th:TH_LOAD_HT"
                     :: "v"(lds_m), "v"(ga_m) : "memory");
    }

    // Zero halos: reproduces jnp.pad(..., (1,1)) boundary semantics.
    if (tid == 0) { s_u[0] = 0.0f;         s_m[0] = 0.0f; }
    if (tid == 1) { s_u[ROW_W + 1] = 0.0f; s_m[ROW_W + 1] = 0.0f; }

    // Wait for this wave's async copies, then barrier so all waves' LDS
    // writes (async + halo ds-stores) are visible.
    asm volatile("s_wait_asynccnt 0" ::: "memory");
    __syncthreads();

    constexpr float Hs     = 0.01f;
    constexpr float INV_H  = 1.0f / Hs;           // compile-time folded
    constexpr float INV_2H = 1.0f / (2.0f * Hs);

    v4f res;
    #pragma unroll
    for (int k = 0; k < 4; ++k) {
        const int j = i + k;            // element index within row
        const float ul = s_u[j];        // u[j-1] (0 at left boundary)
        const float uc = s_u[j + 1];    // u[j]
        const float ur = s_u[j + 2];    // u[j+1] (0 at right boundary)
        const float ml = s_m[j];
        const float mc = s_m[j + 1];
        const float mr = s_m[j + 2];

        // eroded: all three mask cells present
        const float eroded = (ml + mc + mr == 3.0f) ? 1.0f : 0.0f;
        // edge1[i] = (m[i] - m[i-1] == 1)  -> left-sided (forward) diff
        const float edge1  = (mc - ml == 1.0f)  ? 1.0f : 0.0f;
        // edge2[i] = (m[i+1] - m[i] == -1) -> right-sided (backward) diff
        const float edge2  = (mr - mc == -1.0f) ? 1.0f : 0.0f;

        const float internal_d = (ur - ul) * INV_2H;
        const float left_d     = (ur - uc) * INV_H;
        const float right_d    = (uc - ul) * INV_H;

        res[k] = eroded * internal_d + edge1 * left_d + edge2 * right_d;
    }

    // Output is write-once: non-temporal 128-bit store.
    __builtin_nontemporal_store(res, (v4f*)(o_row + i));
}

extern "C" void kernel_launch(void* const* d_in, const int* in_sizes, int n_in,
                              void* d_out, int out_size, void* d_ws, size_t ws_size,
                              hipStream_t stream) {
    const float* u     = (const float*)d_in[0];
    const float* nmask = (const float*)d_in[1];
    float*       out   = (float*)d_out;

    const int rows = in_sizes[0] / ROW_W;   // 8*4*1024 = 32768 rows
    DerivativeNet_kernel<<<rows, 256, 0, stream>>>(u, nmask, out);
}